// FocalLoss_61263413510247
// MI455X (gfx1250) — compile-verified
//
#include <hip/hip_runtime.h>
#include <stdint.h>

// ---------------------------------------------------------------------------
// Focal loss (RetinaNet style), MI455X / gfx1250.
// Bandwidth-bound streaming reduction: 124 MB in -> 4 B out (~5.3 us floor at
// 23.3 TB/s). cls_preds (88 MB, 84 B per anchor) is staged into LDS with
// CDNA5 async LDS loads (global_load_async_to_lds_b128, ASYNCcnt-tracked,
// double buffered); loc tensors use coalesced b128 loads.
// ---------------------------------------------------------------------------

#define TILE        256                   // anchors per tile == blockDim.x
#define NCLS        21
#define TILE_FLOATS (TILE * NCLS)         // 5376 floats
#define TILE_BYTES  (TILE_FLOATS * 4)     // 21504 bytes = 1344 * 16
#define B128_ROUNDS 5                     // 5 * 256 * 16 B = 20480 B
#define TAIL_OFF    20480u                // last 1024 B via per-lane b32

// Issue exactly 6 async-LDS loads per wave (uniform across all 8 waves so the
// s_wait_asynccnt immediates below are exact for every wave).
__device__ __forceinline__ void async_copy_tile(uint32_t ldsByteBase,
                                                const float* __restrict__ gsrc,
                                                int tid) {
  unsigned long long g = (unsigned long long)(uintptr_t)gsrc;
#pragma unroll
  for (int r = 0; r < B128_ROUNDS; ++r) {
    uint32_t off = (uint32_t)(r * TILE + tid) * 16u;
    asm volatile("global_load_async_to_lds_b128 %0, %1, %2"
                 :
                 : "v"(ldsByteBase + off), "v"(off), "s"(g)
                 : "memory");
  }
  uint32_t off2 = TAIL_OFF + (uint32_t)tid * 4u;
  asm volatile("global_load_async_to_lds_b32 %0, %1, %2"
               :
               : "v"(ldsByteBase + off2), "v"(off2), "s"(g)
               : "memory");
}

__global__ __launch_bounds__(TILE) void focal_main_kernel(
    const float4* __restrict__ loc_p,   // [N] of float4
    const float4* __restrict__ loc_t,   // [N] of float4
    const float*  __restrict__ cls_p,   // [N*21]
    const int*    __restrict__ cls_t,   // [N]
    float* __restrict__ acc,            // [3]: locSum, clsSum, numPos
    int numTiles) {
  __shared__ __align__(16) float smem[2 * TILE_FLOATS];  // double-buffered tile
  __shared__ float red[3 * TILE];

  const int tid = threadIdx.x;
  // Generic pointer to LDS: low 32 bits of the shared aperture address are the
  // byte offset from the wave's LDS base, which is what the async-LDS VDST
  // VGPR must contain.
  const uint32_t ldsBase = (uint32_t)(uintptr_t)(&smem[0]);

  float locSum = 0.0f, clsSum = 0.0f, nPos = 0.0f;

  int tile = blockIdx.x;
  const int stride = gridDim.x;
  int buf = 0;

  if (tile < numTiles)
    async_copy_tile(ldsBase, cls_p + (size_t)tile * TILE_FLOATS, tid);

  for (; tile < numTiles; tile += stride) {
    const int nextTile = tile + stride;
    if (nextTile < numTiles) {
      // Prefetch next tile into the other buffer, then wait for the 6 loads of
      // the CURRENT tile (in-order completion: outstanding <= 6 => first 6 done).
      async_copy_tile(ldsBase + (uint32_t)(buf ^ 1) * TILE_BYTES,
                      cls_p + (size_t)nextTile * TILE_FLOATS, tid);
      asm volatile("s_wait_asynccnt 0x6" ::: "memory");
    } else {
      asm volatile("s_wait_asynccnt 0x0" ::: "memory");
    }
    __syncthreads();  // all waves' async writes visible to all lanes

    const int a   = tile * TILE + tid;
    const int tgt = cls_t[a];

    // ---- smooth-L1 over positive anchors (beta = 1, sum reduction) ----
    if (tgt > 0) {
      nPos += 1.0f;
      float4 p = loc_p[a];
      float4 t = loc_t[a];
      float d0 = p.x - t.x, d1 = p.y - t.y, d2 = p.z - t.z, d3 = p.w - t.w;
      float a0 = fabsf(d0), a1 = fabsf(d1), a2 = fabsf(d2), a3 = fabsf(d3);
      locSum += (a0 < 1.0f) ? 0.5f * d0 * d0 : a0 - 0.5f;
      locSum += (a1 < 1.0f) ? 0.5f * d1 * d1 : a1 - 0.5f;
      locSum += (a2 < 1.0f) ? 0.5f * d2 * d2 : a2 - 0.5f;
      locSum += (a3 < 1.0f) ? 0.5f * d3 * d3 : a3 - 0.5f;
    }

    // ---- focal loss over non-ignored anchors ----
    if (tgt > -1) {
      const float* row = &smem[buf * TILE_FLOATS + tid * NCLS];
      float x[NCLS];
#pragma unroll
      for (int c = 0; c < NCLS; ++c) x[c] = row[c];  // conflict-free ds loads
      float m  = x[0];
      float xt = x[0];
#pragma unroll
      for (int c = 1; c < NCLS; ++c) m = fmaxf(m, x[c]);
#pragma unroll
      for (int c = 1; c < NCLS; ++c)
        if (c == tgt) xt = x[c];
      float s = 0.0f;
#pragma unroll
      for (int c = 0; c < NCLS; ++c) s += __expf(x[c] - m);
      float p = __expf(xt - m) / s;
      p = fminf(fmaxf(p, 1e-7f), 1.0f - 1e-7f);
      float q = 1.0f - p;
      clsSum += -__logf(p) * q * q;
    }

    __syncthreads();  // tile fully consumed before next prefetch overwrites it
    buf ^= 1;
  }

  // ---- block reduction of (locSum, clsSum, nPos) ----
  red[tid]            = locSum;
  red[TILE + tid]     = clsSum;
  red[2 * TILE + tid] = nPos;
  __syncthreads();
#pragma unroll
  for (int s = TILE / 2; s > 0; s >>= 1) {
    if (tid < s) {
      red[tid]            += red[tid + s];
      red[TILE + tid]     += red[TILE + tid + s];
      red[2 * TILE + tid] += red[2 * TILE + tid + s];
    }
    __syncthreads();
  }
  if (tid == 0) {
    atomicAdd(&acc[0], red[0]);
    atomicAdd(&acc[1], red[TILE]);
    atomicAdd(&acc[2], red[2 * TILE]);
  }
}

__global__ void focal_zero_kernel(float* acc) {
  if (threadIdx.x < 3) acc[threadIdx.x] = 0.0f;
}

__global__ void focal_finalize_kernel(const float* __restrict__ acc,
                                      float* __restrict__ out) {
  if (threadIdx.x == 0) out[0] = (acc[0] + acc[1]) / acc[2];
}

extern "C" void kernel_launch(void* const* d_in, const int* in_sizes, int n_in,
                              void* d_out, int out_size, void* d_ws,
                              size_t ws_size, hipStream_t stream) {
  (void)n_in; (void)out_size; (void)ws_size;
  const float4* loc_p = (const float4*)d_in[0];  // [B,A,4] f32
  const float4* loc_t = (const float4*)d_in[1];  // [B,A,4] f32
  const float*  cls_p = (const float*)d_in[2];   // [B,A,21] f32
  const int*    cls_t = (const int*)d_in[3];     // [B,A] i32
  // d_in[4] (pos) is recomputed as cls_t > 0; d_in[5] (verbose) ignored.

  float* acc = (float*)d_ws;   // 3 floats of scratch
  float* out = (float*)d_out;  // 1 float

  const int N        = in_sizes[3];      // total anchors (1,048,576)
  const int numTiles = N / TILE;         // 4096 (N is a multiple of 256)
  int grid = numTiles < 1024 ? numTiles : 1024;  // ~4 tiles/block -> pipeline

  focal_zero_kernel<<<dim3(1), dim3(32), 0, stream>>>(acc);
  focal_main_kernel<<<dim3(grid), dim3(TILE), 0, stream>>>(
      loc_p, loc_t, cls_p, cls_t, acc, numTiles);
  focal_finalize_kernel<<<dim3(1), dim3(32), 0, stream>>>(acc, out);
}